// ImprovedRVQ_49228915147311
// MI455X (gfx1250) — compile-verified
//
#include <hip/hip_runtime.h>

typedef __attribute__((ext_vector_type(2))) float v2f;
typedef __attribute__((ext_vector_type(8))) float v8f;

#define B_   16
#define D_   512
#define T_   4096
#define NCB  9
#define K_   1024
#define CD   8
#define TILE 16
#define RSTR 20   // padded LDS row stride (multiple of 4 -> 16B-aligned quads, spreads banks)

// ---------- prep: weight-norm the in-conv rows (len 512), zero-padded to 16 rows ----------
__global__ __launch_bounds__(128) void rvq_prep_in(const float* __restrict__ in_v,
                                                   const float* __restrict__ in_g,
                                                   float* __restrict__ Win) {
  __shared__ float red[128];
  const int row = blockIdx.x;            // 0..143 : i = row>>4, c = row&15
  const int i = row >> 4, c = row & 15;
  const int tid = threadIdx.x;
  float* dst = Win + (size_t)row * D_;
  if (c >= CD) {                          // zero padding rows 8..15
    for (int d = tid; d < D_; d += 128) dst[d] = 0.f;
    return;
  }
  const float* v = in_v + ((size_t)i * CD + c) * D_;
  float s = 0.f;
  for (int d = tid; d < D_; d += 128) { float x = v[d]; s += x * x; }
  red[tid] = s; __syncthreads();
  for (int st = 64; st > 0; st >>= 1) { if (tid < st) red[tid] += red[tid + st]; __syncthreads(); }
  const float scale = in_g[i * CD + c] / sqrtf(red[0]);
  for (int d = tid; d < D_; d += 128) dst[d] = v[d] * scale;
}

// ---------- prep: out-conv rows (len 8) + codebook normalization + zero loss slots ----
__global__ __launch_bounds__(256) void rvq_prep_small(const float* __restrict__ out_v,
                                                      const float* __restrict__ out_g,
                                                      const float* __restrict__ cbs,
                                                      float* __restrict__ Wout,
                                                      float* __restrict__ cn,
                                                      float* __restrict__ loss_slots) {
  const int r = blockIdx.x * blockDim.x + threadIdx.x;
  if (r == 0) { loss_slots[0] = 0.f; loss_slots[1] = 0.f; }
  const int nOut = NCB * D_;
  const int nCb  = NCB * K_;
  if (r < nOut) {
    const float* v = out_v + (size_t)r * CD;
    float s = 0.f;
    #pragma unroll
    for (int c = 0; c < CD; ++c) s += v[c] * v[c];
    const float scale = out_g[r] / sqrtf(s);
    #pragma unroll
    for (int c = 0; c < CD; ++c) Wout[(size_t)r * CD + c] = v[c] * scale;
  } else if (r < nOut + nCb) {
    const int rr = r - nOut;
    const float* v = cbs + (size_t)rr * CD;
    float s = 0.f;
    #pragma unroll
    for (int c = 0; c < CD; ++c) s += v[c] * v[c];
    const float scale = 1.f / fmaxf(sqrtf(s), 1e-12f);
    #pragma unroll
    for (int c = 0; c < CD; ++c) cn[(size_t)rr * CD + c] = v[c] * scale;
  }
}

// ---------- fused main: all 9 RVQ stages, residual tile resident in LDS ----------
__global__ __launch_bounds__(256) void rvq_main(const float* __restrict__ z,
                                                const float* __restrict__ in_b,
                                                const float* __restrict__ out_b,
                                                const float* __restrict__ cbs,   // raw codebooks
                                                const float* __restrict__ Win,   // [9][16][512] padded
                                                const float* __restrict__ Wout,  // [9][512][8]
                                                const float* __restrict__ cn,    // [9][1024][8]
                                                float* __restrict__ out_zq,
                                                float* __restrict__ out_codes,
                                                float* __restrict__ out_loss) {
  __shared__ float R[D_ * RSTR];          // residual tile [512][16] (padded) ~40KB
  __shared__ float zePart[8][CD][TILE];   // per-wave partial z_e
  __shared__ float ze[CD][TILE];
  __shared__ float zeq[CD][TILE];
  __shared__ float bout[D_];
  __shared__ float part[256];
  __shared__ float bestV[8][32];
  __shared__ int   bestI[8][32];
  __shared__ int   colIdx[TILE];

  const int blk  = blockIdx.x;
  const int bidx = blk / (T_ / TILE);
  const int t0   = (blk % (T_ / TILE)) * TILE;
  const int tid  = threadIdx.x;
  const int lane = tid & 31;
  const int wave = tid >> 5;
  const int colB = lane & 15;             // column (B/C fragment layout)
  const int hi   = lane >> 4;             // half-wave selector

  const float* zb = z + (size_t)bidx * D_ * T_;

  // ---- async DMA of the residual tile into LDS (ASYNCcnt path, bypasses VGPRs) ----
  #pragma unroll
  for (int q = 0; q < 8; ++q) {
    const int g = tid + 256 * q;                 // 2048 float4 groups
    const int d = g >> 2, quad = g & 3;
    const float* src = zb + (size_t)d * T_ + t0 + quad * 4;
    const unsigned int dst = (unsigned int)(uintptr_t)(&R[d * RSTR + quad * 4]);
    asm volatile("global_load_async_to_lds_b128 %0, %1, off"
                 :: "v"(dst), "v"(src) : "memory");
  }
  asm volatile("s_wait_asynccnt 0x0" ::: "memory");
  float lossAcc = 0.f;
  __syncthreads();

  for (int i = 0; i < NCB; ++i) {
    // stage out-bias into LDS (used by zq phase after the barriers below)
    bout[tid]       = out_b[i * D_ + tid];
    bout[tid + 256] = out_b[i * D_ + tid + 256];

    // ---- z_e = Win[i](16x512, rows 8..15 zero) @ R(512x16) via WMMA, K split over waves ----
    {
      const float* Wi = Win + (size_t)i * 16 * D_;
      const int m = colB;                         // A row = lane%16
      v8f zacc = {};
      #pragma unroll
      for (int kc = 0; kc < 16; ++kc) {
        const int k = wave * 64 + kc * 4;
        const v2f a = *(const v2f*)(Wi + (size_t)m * D_ + k + hi * 2);
        v2f bfr;
        bfr.x = R[(k + hi * 2 + 0) * RSTR + colB];
        bfr.y = R[(k + hi * 2 + 1) * RSTR + colB];
        zacc = __builtin_amdgcn_wmma_f32_16x16x4_f32(false, a, false, bfr, (short)0, zacc, false, false);
      }
      if (lane < 16) {                            // rows 0..7 live in lanes 0..15, VGPRs 0..7
        #pragma unroll
        for (int v = 0; v < 8; ++v) zePart[wave][v][lane] = zacc[v];
      }
    }
    __syncthreads();
    if (tid < 128) {
      const int col = tid & 15, cc = tid >> 4;
      float s = in_b[i * CD + cc];
      #pragma unroll
      for (int w = 0; w < 8; ++w) s += zePart[w][cc][col];
      ze[cc][col] = s;
    }
    __syncthreads();

    // ---- sim = c_n[1024x8] @ z_e[8x16] via WMMA, fused argmax ----
    v2f bf0, bf1;
    bf0.x = ze[hi * 2 + 0][colB];
    bf0.y = ze[hi * 2 + 1][colB];
    bf1.x = ze[4 + hi * 2 + 0][colB];
    bf1.y = ze[4 + hi * 2 + 1][colB];

    float bestv = -3.0e38f;
    int   besti = 0;
    const float* cni = cn + (size_t)i * K_ * CD;
    for (int jb = 0; jb < 8; ++jb) {              // each wave: 8 row-blocks of 16
      const int kb = wave * 8 + jb;
      const float* crow = cni + (size_t)(kb * 16 + colB) * CD;
      const int k0 = hi * 2;
      const v2f a0 = *(const v2f*)(crow + k0);
      const v2f a1 = *(const v2f*)(crow + 4 + k0);
      v8f acc = {};
      acc = __builtin_amdgcn_wmma_f32_16x16x4_f32(false, a0, false, bf0, (short)0, acc, false, false);
      acc = __builtin_amdgcn_wmma_f32_16x16x4_f32(false, a1, false, bf1, (short)0, acc, false, false);
      const int rbase = kb * 16 + hi * 8;
      #pragma unroll
      for (int v = 0; v < 8; ++v) {
        const float val = acc[v];
        if (val > bestv) { bestv = val; besti = rbase + v; }
      }
    }
    bestV[wave][lane] = bestv;
    bestI[wave][lane] = besti;
    __syncthreads();

    if (tid < TILE) {                             // final argmax per column
      float bv = -3.0e38f; int bi2 = 0x7fffffff;
      for (int w = 0; w < 8; ++w)
        #pragma unroll
        for (int h = 0; h < 2; ++h) {
          const float v = bestV[w][tid + h * 16];
          const int  ii = bestI[w][tid + h * 16];
          if (v > bv || (v == bv && ii < bi2)) { bv = v; bi2 = ii; }
        }
      colIdx[tid] = bi2;
      out_codes[((size_t)(bidx * T_ + t0 + tid)) * NCB + i] = (float)bi2;
    }
    __syncthreads();

    // ---- gather raw codebook rows ----
    if (tid < CD * TILE) {
      const int c = tid & 7, col = tid >> 3;
      zeq[c][col] = cbs[((size_t)i * K_ + colIdx[col]) * CD + c];
    }
    __syncthreads();

    // ---- losses (commit == cb_loss in forward) ----
    if (tid < CD * TILE) {
      const int c = tid & 7, col = tid >> 3;
      const float dlt = ze[c][col] - zeq[c][col];
      lossAcc += dlt * dlt;
    }

    // ---- zq_i = Wout[i](512x8) @ z_eq(8x16) + b_out via WMMA ; R -= zq_i in place ----
    {
      v2f qb0, qb1;
      qb0.x = zeq[hi * 2 + 0][colB];
      qb0.y = zeq[hi * 2 + 1][colB];
      qb1.x = zeq[4 + hi * 2 + 0][colB];
      qb1.y = zeq[4 + hi * 2 + 1][colB];
      #pragma unroll
      for (int rb = 0; rb < 4; ++rb) {
        const int blkrow = wave * 4 + rb;         // 32 row-blocks of 16
        const float* wrow = Wout + ((size_t)i * D_ + blkrow * 16 + colB) * CD;
        const int k0 = hi * 2;
        const v2f a0 = *(const v2f*)(wrow + k0);
        const v2f a1 = *(const v2f*)(wrow + 4 + k0);
        v8f acc = {};
        acc = __builtin_amdgcn_wmma_f32_16x16x4_f32(false, a0, false, qb0, (short)0, acc, false, false);
        acc = __builtin_amdgcn_wmma_f32_16x16x4_f32(false, a1, false, qb1, (short)0, acc, false, false);
        const int rbase = blkrow * 16 + hi * 8;   // each (row,col) owned by exactly one lane
        #pragma unroll
        for (int v = 0; v < 8; ++v) {
          const int row = rbase + v;
          R[row * RSTR + colB] -= acc[v] + bout[row];
        }
      }
    }
    __syncthreads();
  }

  // ---- z_q = z - residual_final ----
  for (int e = tid; e < D_ * TILE; e += 256) {
    const int d = e / TILE, c = e % TILE;
    const size_t off = ((size_t)bidx * D_ + d) * T_ + t0 + c;
    out_zq[off] = zb[(size_t)d * T_ + t0 + c] - R[d * RSTR + c];
  }

  // ---- block loss reduce + atomic accumulate into both scalar slots ----
  part[tid] = lossAcc;
  __syncthreads();
  for (int s = 128; s > 0; s >>= 1) { if (tid < s) part[tid] += part[tid + s]; __syncthreads(); }
  if (tid == 0) {
    const float v = part[0] * (1.0f / (float)((size_t)B_ * CD * T_));
    atomicAdd(&out_loss[0], v);
    atomicAdd(&out_loss[1], v);
  }
}

extern "C" void kernel_launch(void* const* d_in, const int* in_sizes, int n_in,
                              void* d_out, int out_size, void* d_ws, size_t ws_size,
                              hipStream_t stream) {
  const float* z     = (const float*)d_in[0];
  const float* in_v  = (const float*)d_in[1];
  const float* in_g  = (const float*)d_in[2];
  const float* in_b  = (const float*)d_in[3];
  const float* out_v = (const float*)d_in[4];
  const float* out_g = (const float*)d_in[5];
  const float* out_b = (const float*)d_in[6];
  const float* cbs   = (const float*)d_in[7];

  float* ws   = (float*)d_ws;
  float* Win  = ws;                               // [9][16][512] padded = 73728
  float* Wout = Win + (size_t)NCB * 16 * D_;      // [9][512][8]         = 36864
  float* cn   = Wout + (size_t)NCB * D_ * CD;     // [9][1024][8]        = 73728

  float* out   = (float*)d_out;
  float* zq    = out;
  float* codes = zq + (size_t)B_ * D_ * T_;
  float* loss  = codes + (size_t)B_ * T_ * NCB;

  rvq_prep_in<<<NCB * 16, 128, 0, stream>>>(in_v, in_g, Win);
  const int nSmall = NCB * D_ + NCB * K_;
  rvq_prep_small<<<(nSmall + 255) / 256, 256, 0, stream>>>(out_v, out_g, cbs, Wout, cn, loss);
  rvq_main<<<B_ * (T_ / TILE), 256, 0, stream>>>(z, in_b, out_b, cbs, Win, Wout, cn,
                                                 zq, codes, loss);
}